// Attention_Gate_8211977470570
// MI455X (gfx1250) — compile-verified
//
#include <hip/hip_runtime.h>
#include <hip/hip_bf16.h>
#include <math.h>

typedef __attribute__((ext_vector_type(2))) float v2f;
typedef __attribute__((ext_vector_type(8))) float v8f;

__global__ __launch_bounds__(256) void zero_kernel(float* __restrict__ p, long n) {
  long i = (long)blockIdx.x * blockDim.x + threadIdx.x;
  long stride = (long)gridDim.x * blockDim.x;
  for (; i < n; i += stride) p[i] = 0.f;
}

// Gather-GEMM-scatter sparse conv: out[dst[k,e], :64] += x[src[k,e], :CIN] @ W[k]
// One wave32 per (k = blockIdx.y, 16-edge tile). fp32 WMMA 16x16x4, 4 N-tiles.
// Out-of-range edges: gather index is clamped (D row M depends only on A row M,
// and those D rows are skipped at scatter), so no masking needed in the K-loop.
template<int CIN>
__global__ __launch_bounds__(256) void sconv_wmma(
    const float* __restrict__ x, const float* __restrict__ W,
    const int* __restrict__ src, const int* __restrict__ dst,
    float* __restrict__ out, int E, int tilesPerK)
{
  const int t = blockIdx.x * 8 + (threadIdx.x >> 5);  // edge tile
  if (t >= tilesPerK) return;
  const int k = blockIdx.y;
  const int lane = threadIdx.x & 31;
  const int g = lane >> 4;        // half-wave group
  const int n16 = lane & 15;      // M row for A gather / N col for B & D
  const int ebase = t << 4;

  // A-matrix gather (f32 A 16x4 layout: lanes 0-15 K=0,1; lanes 16-31 K=2,3)
  int e = ebase + n16;
  if (e >= E) e = E - 1;                       // clamp; result discarded at scatter
  const float* xr = x + (size_t)src[(size_t)k * E + e] * CIN;
  const float* Wk = W + (size_t)k * CIN * 64;

  v8f acc[4];
#pragma unroll
  for (int n = 0; n < 4; ++n)
#pragma unroll
    for (int j = 0; j < 8; ++j) acc[n][j] = 0.f;

#pragma unroll
  for (int s = 0; s < CIN / 4; ++s) {
    const int cb = 4 * s + 2 * g;              // this lane's 2 K-channels
    v2f a = *(const v2f*)(xr + cb);
    const float* wp = Wk + (size_t)cb * 64 + n16;  // B rows cb, cb+1; col n*16+n16
#pragma unroll
    for (int n = 0; n < 4; ++n) {
      v2f b;
      b.x = wp[n * 16];
      b.y = wp[n * 16 + 64];
      acc[n] = __builtin_amdgcn_wmma_f32_16x16x4_f32(
          false, a, false, b, (short)0, acc[n], false, false);
    }
  }

  // D layout: VGPR j -> M = j + 8*g, N = n16. Scatter-add per element.
  int drow[8];
#pragma unroll
  for (int j = 0; j < 8; ++j) {
    int ed = ebase + j + 8 * g;
    drow[j] = (ed < E) ? dst[(size_t)k * E + ed] : -1;
  }
#pragma unroll
  for (int n = 0; n < 4; ++n)
#pragma unroll
    for (int j = 0; j < 8; ++j)
      if (drow[j] >= 0)
        atomicAdd(out + (size_t)drow[j] * 64 + n * 16 + n16, acc[n][j]);
}

// Per-channel sum & sum-of-squares over N rows of [N,64]: LDS ds_add_f32
// block reduction, then one global atomic per channel per block.
__global__ __launch_bounds__(256) void stats64(
    const float* __restrict__ x, int N, float* __restrict__ st /*[128]*/)
{
  __shared__ float ls[64], lq[64];
  int tid = threadIdx.x;
  if (tid < 64) { ls[tid] = 0.f; lq[tid] = 0.f; }
  __syncthreads();
  int c = tid & 63;
  int gt = blockIdx.x * blockDim.x + tid;
  int rstart = gt >> 6;
  int rstride = (gridDim.x * blockDim.x) >> 6;
  float s = 0.f, q = 0.f;
  for (int r = rstart; r < N; r += rstride) {
    float v = x[(size_t)r * 64 + c];
    s += v; q += v * v;
  }
  atomicAdd(&ls[c], s);
  atomicAdd(&lq[c], q);
  __syncthreads();
  if (tid < 64) {
    atomicAdd(&st[tid], ls[tid]);
    atomicAdd(&st[64 + tid], lq[tid]);
  }
}

// In-place BN(train) + LeakyReLU; optional fused "+ addend" (for s = phi + theta)
__global__ __launch_bounds__(256) void bn_leaky64(
    float* __restrict__ x, int N, const float* __restrict__ st,
    const float* __restrict__ gam, const float* __restrict__ bet,
    const float* __restrict__ addend)
{
  long total = (long)N * 64;
  long stride = (long)gridDim.x * blockDim.x;
  for (long i = (long)blockIdx.x * blockDim.x + threadIdx.x; i < total; i += stride) {
    int c = (int)(i & 63);
    float m = st[c] / N;
    float v = st[64 + c] / N - m * m;
    float r = rsqrtf(v + 1e-5f);
    float y = (x[i] - m) * r * gam[c] + bet[c];
    y = (y >= 0.f) ? y : 0.01f * y;
    if (addend) y += addend[i];
    x[i] = y;
  }
}

__global__ __launch_bounds__(256) void bn_sigmoid64(
    float* __restrict__ x, int N, const float* __restrict__ st,
    const float* __restrict__ gam, const float* __restrict__ bet)
{
  long total = (long)N * 64;
  long stride = (long)gridDim.x * blockDim.x;
  for (long i = (long)blockIdx.x * blockDim.x + threadIdx.x; i < total; i += stride) {
    int c = (int)(i & 63);
    float m = st[c] / N;
    float v = st[64 + c] / N - m * m;
    float r = rsqrtf(v + 1e-5f);
    float y = (x[i] - m) * r * gam[c] + bet[c];
    x[i] = 1.f / (1.f + __expf(-y));
  }
}

// Cout=1 sparse conv: out[dst[e]] += dot(x[src[e], :64], Wsum[k, :64]); k = blockIdx.y
__global__ __launch_bounds__(256) void sconv_sum(
    const float* __restrict__ x, const float* __restrict__ Wsum,
    const int* __restrict__ src, const int* __restrict__ dst,
    float* __restrict__ out, int E)
{
  int i = blockIdx.x * blockDim.x + threadIdx.x;
  if (i >= E) return;
  int k = blockIdx.y;
  size_t idx = (size_t)k * E + i;
  const float4* row = (const float4*)(x + (size_t)src[idx] * 64);
  const float4* w = (const float4*)(Wsum + (size_t)k * 64);
  float acc = 0.f;
#pragma unroll
  for (int c = 0; c < 16; ++c) {
    float4 a = row[c], b = w[c];
    acc += a.x * b.x + a.y * b.y + a.z * b.z + a.w * b.w;
  }
  atomicAdd(&out[dst[idx]], acc);
}

// Scalar sum & sumsq over N values
__global__ __launch_bounds__(256) void stats1(
    const float* __restrict__ x, int N, float* __restrict__ st /*[2]*/)
{
  __shared__ float ss[2];
  if (threadIdx.x == 0) { ss[0] = 0.f; ss[1] = 0.f; }
  __syncthreads();
  float s = 0.f, q = 0.f;
  for (int i = blockIdx.x * blockDim.x + threadIdx.x; i < N;
       i += gridDim.x * blockDim.x) {
    float v = x[i]; s += v; q += v * v;
  }
#pragma unroll
  for (int o = 16; o > 0; o >>= 1) {
    s += __shfl_down(s, o, 32);
    q += __shfl_down(q, o, 32);
  }
  if ((threadIdx.x & 31) == 0) { atomicAdd(&ss[0], s); atomicAdd(&ss[1], q); }
  __syncthreads();
  if (threadIdx.x == 0) { atomicAdd(&st[0], ss[0]); atomicAdd(&st[1], ss[1]); }
}

__global__ __launch_bounds__(256) void bn_leaky1(
    float* __restrict__ x, int N, const float* __restrict__ st,
    const float* __restrict__ gam, const float* __restrict__ bet)
{
  int i = blockIdx.x * blockDim.x + threadIdx.x;
  if (i >= N) return;
  float m = st[0] / N;
  float v = st[1] / N - m * m;
  float r = rsqrtf(v + 1e-5f);
  float y = (x[i] - m) * r * gam[0] + bet[0];
  x[i] = (y >= 0.f) ? y : 0.01f * y;
}

// Cin=1 inverse conv: out[sidx[e], :64] += ssum[gidx[e]] * Wup[k, :64]
// 2 threads per edge, 32 channels each; k = blockIdx.y
__global__ __launch_bounds__(256) void sconv_up(
    const float* __restrict__ ssum, const float* __restrict__ Wup,
    const int* __restrict__ gidx, const int* __restrict__ sidx,
    float* __restrict__ out, int E)
{
  int t = blockIdx.x * blockDim.x + threadIdx.x;
  if (t >= E * 2) return;
  int e = t >> 1;
  int half = t & 1;
  int k = blockIdx.y;
  size_t idx = (size_t)k * E + e;
  float val = ssum[gidx[idx]];
  const float* w = Wup + (size_t)k * 64 + half * 32;
  float* o = out + (size_t)sidx[idx] * 64 + half * 32;
#pragma unroll
  for (int c = 0; c < 32; ++c) atomicAdd(o + c, val * w[c]);
}

extern "C" void kernel_launch(void* const* d_in, const int* in_sizes, int n_in,
                              void* d_out, int out_size, void* d_ws, size_t ws_size,
                              hipStream_t stream) {
  const float* gate     = (const float*)d_in[0];
  const float* shortcut = (const float*)d_in[1];
  const int* src_down   = (const int*)d_in[2];
  const int* dst_down   = (const int*)d_in[3];
  const int* src_g      = (const int*)d_in[4];
  const int* dst_g      = (const int*)d_in[5];
  const float* W_sc  = (const float*)d_in[6];
  const float* W_g   = (const float*)d_in[8];
  const float* W_gu  = (const float*)d_in[10];
  const float* W_sum = (const float*)d_in[12];
  const float* W_up  = (const float*)d_in[13];
  // biases (d_in[7,9,11,14]) cancel exactly inside training-mode BatchNorm.
  const float* gam_sc  = (const float*)d_in[15]; const float* bet_sc  = (const float*)d_in[16];
  const float* gam_g   = (const float*)d_in[17]; const float* bet_g   = (const float*)d_in[18];
  const float* gam_gu  = (const float*)d_in[19]; const float* bet_gu  = (const float*)d_in[20];
  const float* gam_sum = (const float*)d_in[21]; const float* bet_sum = (const float*)d_in[22];
  const float* gam_up  = (const float*)d_in[23]; const float* bet_up  = (const float*)d_in[24];

  const int K = 27;
  const int Nc  = in_sizes[0] / 128;
  const int Nf  = in_sizes[1] / 64;
  const int Ekd = in_sizes[2] / K;
  const int Ekg = in_sizes[4] / K;
  const size_t nc64 = (size_t)Nc * 64;

  // Workspace layout; fall back to staging theta/phi in d_out (dead until the
  // final scatter) if the scratch buffer is small.
  float* theta; float* phi; float* sbuf; float* ssum; float* st;
  size_t need_full = (3 * nc64 + (size_t)Nc + 1024) * sizeof(float);
  if (ws_size >= need_full) {
    float* w = (float*)d_ws;
    theta = w; phi = w + nc64; sbuf = w + 2 * nc64;
    ssum = w + 3 * nc64; st = ssum + Nc;
  } else {
    theta = (float*)d_out; phi = theta + nc64;
    float* w = (float*)d_ws;
    sbuf = w; ssum = w + nc64; st = ssum + Nc;
  }
  // stats: [0..127]=sc, [128..255]=g, [256..383]=gu, [384..385]=sum, [512..639]=up

  const int tpkD = (Ekd + 15) / 16;
  const int tpkG = (Ekg + 15) / 16;
  const dim3 gD((tpkD + 7) / 8, K), gG((tpkG + 7) / 8, K);
  const int ZB = 2048, EB = 4096;

  // zero accumulators + stats (theta..phi and sbuf..st are contiguous in both layouts)
  zero_kernel<<<ZB, 256, 0, stream>>>(theta, (long)(2 * nc64));
  zero_kernel<<<ZB, 256, 0, stream>>>(sbuf, (long)(nc64 + Nc + 1024));

  // theta = sconv(shortcut, W_sc) ; phi = sconv(gate, W_g)
  sconv_wmma<64><<<gD, 256, 0, stream>>>(
      shortcut, W_sc, src_down, dst_down, theta, Ekd, tpkD);
  sconv_wmma<128><<<gG, 256, 0, stream>>>(
      gate, W_g, src_g, dst_g, phi, Ekg, tpkG);

  stats64<<<256, 256, 0, stream>>>(theta, Nc, st + 0);
  stats64<<<256, 256, 0, stream>>>(phi, Nc, st + 128);
  bn_leaky64<<<EB, 256, 0, stream>>>(theta, Nc, st + 0, gam_sc, bet_sc, nullptr);
  bn_leaky64<<<EB, 256, 0, stream>>>(phi, Nc, st + 128, gam_g, bet_g, nullptr);

  // phi2 = sconv(phi, W_gu) with transposed index pairs
  sconv_wmma<64><<<gG, 256, 0, stream>>>(
      phi, W_gu, dst_g, src_g, sbuf, Ekg, tpkG);
  stats64<<<256, 256, 0, stream>>>(sbuf, Nc, st + 256);
  // s = leaky(bn(phi2)) + theta (fused)
  bn_leaky64<<<EB, 256, 0, stream>>>(sbuf, Nc, st + 256, gam_gu, bet_gu, theta);

  // 1-channel submanifold conv + BN + leaky
  sconv_sum<<<dim3((Ekg + 255) / 256, K), 256, 0, stream>>>(
      sbuf, W_sum, src_g, dst_g, ssum, Ekg);
  stats1<<<128, 256, 0, stream>>>(ssum, Nc, st + 384);
  bn_leaky1<<<(Nc + 255) / 256, 256, 0, stream>>>(ssum, Nc, st + 384, gam_sum, bet_sum);

  // inverse conv to fine resolution -> d_out, then BN + sigmoid in place
  zero_kernel<<<ZB, 256, 0, stream>>>((float*)d_out, (long)Nf * 64);
  sconv_up<<<dim3((2 * Ekd + 255) / 256, K), 256, 0, stream>>>(
      ssum, W_up, dst_down, src_down, (float*)d_out, Ekd);
  stats64<<<512, 256, 0, stream>>>((float*)d_out, Nf, st + 512);
  bn_sigmoid64<<<EB, 256, 0, stream>>>((float*)d_out, Nf, st + 512, gam_up, bet_up);
}